// GNN_19868518711604
// MI455X (gfx1250) — compile-verified
//
#include <hip/hip_runtime.h>

typedef __attribute__((ext_vector_type(2))) float v2f;
typedef __attribute__((ext_vector_type(8))) float v8f;

// ---------------------------------------------------------------------------
// K1: zero agg1 [8N], set deg[i] = 1.0 (self-loop weight contribution)
// ---------------------------------------------------------------------------
__global__ void k_init(float* __restrict__ deg, float* __restrict__ agg1, int n) {
    int i = blockIdx.x * blockDim.x + threadIdx.x;
    if (i < 8 * n) agg1[i] = 0.0f;
    if (i < n)     deg[i]  = 1.0f;
}

// K2: deg[col[e]] += w[e]   (scatter-add, L2-resident target)
__global__ void k_deg(const long long* __restrict__ col, const float* __restrict__ w,
                      float* __restrict__ deg, int e) {
    int i = blockIdx.x * blockDim.x + threadIdx.x;
    if (i < e) atomicAdd(&deg[(int)col[i]], w[i]);
}

// K3: deg -> dinv = rsqrt(deg) (in place)
__global__ void k_dinv(float* __restrict__ dv, int n) {
    int i = blockIdx.x * blockDim.x + threadIdx.x;
    if (i < n) {
        float d = dv[i];
        dv[i] = (d > 0.0f) ? rsqrtf(d) : 0.0f;
    }
}

// K4: norm[e] = dinv[row]*w*dinv[col]  (computed once, reused by both layers)
__global__ void k_norm(const long long* __restrict__ row, const long long* __restrict__ col,
                       const float* __restrict__ w, const float* __restrict__ dv,
                       float* __restrict__ norm, int e) {
    int i = blockIdx.x * blockDim.x + threadIdx.x;
    if (i < e) norm[i] = dv[(int)row[i]] * w[i] * dv[(int)col[i]];
}

// K5: layer-1 aggregation in the *input* (8-wide) space: agg1[col,f] += norm*x[row,f]
// one thread per (edge, feature): coalesced gathers and 8-consecutive atomics.
__global__ void k_agg1(const long long* __restrict__ row, const long long* __restrict__ col,
                       const float* __restrict__ norm, const float* __restrict__ x,
                       float* __restrict__ agg1, int e) {
    long long t = (long long)blockIdx.x * blockDim.x + threadIdx.x;
    if (t >= 8LL * e) return;
    int ei = (int)(t >> 3);
    int f  = (int)(t & 7);
    int r  = (int)row[ei];
    int c  = (int)col[ei];
    atomicAdd(&agg1[c * 8 + f], norm[ei] * x[r * 8 + f]);
}

// ---------------------------------------------------------------------------
// K6: per 16-node tile (one wave32 each):
//   A[m,k] = agg1[m,k] + dinv[m]^2 * x[m,k]      (fold in self-loop)
//   H = A @ W1  via v_wmma_f32_16x16x4_f32  (K=8 -> 2 wmma steps, 4 N-tiles)
//   z[m] = sum_c relu(H[m,c] + b1[c]) * W2[c]    (fused, H never stored)
// ---------------------------------------------------------------------------
__global__ void k_gemm_z(const float* __restrict__ agg1, const float* __restrict__ x,
                         const float* __restrict__ dv,   const float* __restrict__ W1,
                         const float* __restrict__ bias1, const float* __restrict__ W2,
                         float* __restrict__ z, int n, int ntiles) {
    int wave = (blockIdx.x * blockDim.x + threadIdx.x) >> 5;
    int lane = threadIdx.x & 31;
    if (wave >= ntiles) return;            // wave-uniform: EXEC stays all-1s inside

    int base = wave * 16;
    int nl   = lane & 15;                  // column-in-tile / row-in-tile index
    int half = lane >> 4;                  // 0: lanes 0-15, 1: lanes 16-31

    // --- A fragments (ISA 32-bit A 16x4 layout): M = nl, K split by lane half ---
    int m = base + nl;
    if (m >= n) m = n - 1;                 // clamp (guard on store below)
    float di = dv[m];
    float s  = di * di;                    // self-loop norm dinv^2 * 1.0
    int k0 = 2 * half;                     // lanes 0-15: K={0,1}, lanes 16-31: K={2,3}
    v2f a0, a1;
    a0.x = agg1[m * 8 + k0]     + s * x[m * 8 + k0];
    a0.y = agg1[m * 8 + k0 + 1] + s * x[m * 8 + k0 + 1];
    a1.x = agg1[m * 8 + k0 + 4] + s * x[m * 8 + k0 + 4];   // second K-step (k+4)
    a1.y = agg1[m * 8 + k0 + 5] + s * x[m * 8 + k0 + 5];

    float zp[8];
#pragma unroll
    for (int r = 0; r < 8; ++r) zp[r] = 0.0f;

#pragma unroll
    for (int t = 0; t < 4; ++t) {          // 4 tiles over HID=64
        int cn = t * 16 + nl;              // output column
        // B fragments (4x16, row-striped like C/D): K = vgpr + 2*half
        v2f b0, b1v;
        b0.x  = W1[(k0)     * 64 + cn];
        b0.y  = W1[(k0 + 1) * 64 + cn];
        b1v.x = W1[(k0 + 4) * 64 + cn];
        b1v.y = W1[(k0 + 5) * 64 + cn];

        v8f c = {};
        c = __builtin_amdgcn_wmma_f32_16x16x4_f32(false, a0, false, b0,  (short)0, c, false, false);
        c = __builtin_amdgcn_wmma_f32_16x16x4_f32(false, a1, false, b1v, (short)0, c, false, false);

        // C/D layout: lane holds column cn; VGPR r -> row M = r + 8*half
        float bv  = bias1[cn];
        float w2c = W2[cn];
#pragma unroll
        for (int r = 0; r < 8; ++r) {
            float hv = c[r] + bv;
            hv = hv > 0.0f ? hv : 0.0f;    // relu
            zp[r] += hv * w2c;             // partial dot with W2 (this lane's column)
        }
    }

    // reduce partials across the 16 lanes of each half (xor masks stay in-half)
#pragma unroll
    for (int r = 0; r < 8; ++r) {
        float v = zp[r];
        v += __shfl_xor(v, 1, 32);
        v += __shfl_xor(v, 2, 32);
        v += __shfl_xor(v, 4, 32);
        v += __shfl_xor(v, 8, 32);
        zp[r] = v;
    }
    if (nl == 0) {                         // lane 0 -> rows 0..7, lane 16 -> rows 8..15
#pragma unroll
        for (int r = 0; r < 8; ++r) {
            int mm = base + half * 8 + r;
            if (mm < n) z[mm] = zp[r];
        }
    }
}

// K7: out[i] = b2 + dinv[i]^2 * z[i]   (bias + layer-2 self-loop term)
__global__ void k_out_init(const float* __restrict__ dv, const float* __restrict__ z,
                           const float* __restrict__ b2, float* __restrict__ out, int n) {
    int i = blockIdx.x * blockDim.x + threadIdx.x;
    if (i < n) out[i] = b2[0] + dv[i] * dv[i] * z[i];
}

// K8: layer-2 edge aggregation on scalars: out[col] += norm * z[row]
__global__ void k_agg2(const long long* __restrict__ row, const long long* __restrict__ col,
                       const float* __restrict__ norm, const float* __restrict__ z,
                       float* __restrict__ out, int e) {
    int i = blockIdx.x * blockDim.x + threadIdx.x;
    if (i < e) atomicAdd(&out[(int)col[i]], norm[i] * z[(int)row[i]]);
}

// ---------------------------------------------------------------------------
extern "C" void kernel_launch(void* const* d_in, const int* in_sizes, int n_in,
                              void* d_out, int out_size, void* d_ws, size_t ws_size,
                              hipStream_t stream) {
    const float*     x   = (const float*)d_in[0];       // [N,8]
    const long long* ei  = (const long long*)d_in[1];   // [2,E] int64
    const float*     w   = (const float*)d_in[2];       // [E]
    const float*     W1  = (const float*)d_in[3];       // [8,64]
    const float*     b1  = (const float*)d_in[4];       // [64]
    const float*     W2  = (const float*)d_in[5];       // [64,1]
    const float*     b2  = (const float*)d_in[6];       // [1]

    const int n = in_sizes[0] / 8;
    const int e = in_sizes[2];
    const long long* row = ei;
    const long long* col = ei + e;

    // workspace layout (floats): dinv[n] | norm[e] | agg1[8n] | z[n]
    float* dv   = (float*)d_ws;
    float* norm = dv + n;
    float* agg1 = norm + e;
    float* z    = agg1 + (size_t)8 * n;
    float* out  = (float*)d_out;

    const int thr = 256;
    k_init <<<(8 * n + thr - 1) / thr, thr, 0, stream>>>(dv, agg1, n);
    k_deg  <<<(e + thr - 1) / thr, thr, 0, stream>>>(col, w, dv, e);
    k_dinv <<<(n + thr - 1) / thr, thr, 0, stream>>>(dv, n);
    k_norm <<<(e + thr - 1) / thr, thr, 0, stream>>>(row, col, w, dv, norm, e);

    long long tot = 8LL * e;
    k_agg1 <<<(int)((tot + thr - 1) / thr), thr, 0, stream>>>(row, col, norm, x, agg1, e);

    const int ntiles = (n + 15) / 16;
    const int wpb    = thr / 32;           // 8 waves per block
    k_gemm_z<<<(ntiles + wpb - 1) / wpb, thr, 0, stream>>>(agg1, x, dv, W1, b1, W2, z, n, ntiles);

    k_out_init<<<(n + thr - 1) / thr, thr, 0, stream>>>(dv, z, b2, out, n);
    k_agg2    <<<(e + thr - 1) / thr, thr, 0, stream>>>(row, col, norm, z, out, e);
}